// IterativeLearnToTrustV2_83794811945384
// MI455X (gfx1250) — compile-verified
//
#include <hip/hip_runtime.h>
#include <hip/hip_bf16.h>

typedef __attribute__((ext_vector_type(2))) float v2f;
typedef __attribute__((ext_vector_type(8))) float v8f;

#define ALPHA_C    0.999f
#define ONE_MALPHA 0.001f
#define NUM_ITER   50
#define NF_CONST   25000   // reference constant: frozen block follows learnable block

// ---------------------------------------------------------------------------
// One-time: build H[n,c] = w_n * onehot(argmax(pred[n,:])) and Z0 = H.
// ---------------------------------------------------------------------------
__global__ __launch_bounds__(256) void build_h_kernel(
    const float* __restrict__ pred, const float* __restrict__ margins,
    const float* __restrict__ raw, float* __restrict__ H, float* __restrict__ Z0,
    int n_nodes, int n_learn, int n_frozen) {
  int n = blockIdx.x * blockDim.x + threadIdx.x;
  if (n >= n_nodes) return;

  float w;
  if (n < n_learn)                 w = 1.0f / (1.0f + __expf(-raw[n]));   // sigmoid
  else if (n < n_learn + n_frozen) w = 1.0f;
  else                             w = 0.0f;
  w *= margins[n];

  const float4* p4 = (const float4*)(pred + (size_t)n * 16);
  float best = -3.402823466e38f;
  int bi = 0;
#pragma unroll
  for (int q = 0; q < 4; ++q) {
    float4 v = p4[q];
    float vv[4] = {v.x, v.y, v.z, v.w};
#pragma unroll
    for (int j = 0; j < 4; ++j) {
      int idx = q * 4 + j;
      if (vv[j] > best) { best = vv[j]; bi = idx; }   // first-max == jnp.argmax
    }
  }

  float4 o[4];
  float* of = (float*)o;
#pragma unroll
  for (int c = 0; c < 16; ++c) of[c] = (c == bi) ? w : 0.0f;

  float4* h4 = (float4*)(H  + (size_t)n * 16);
  float4* z4 = (float4*)(Z0 + (size_t)n * 16);
#pragma unroll
  for (int q = 0; q < 4; ++q) { h4[q] = o[q]; z4[q] = o[q]; }
}

// ---------------------------------------------------------------------------
// Per-iteration init: Zdst(16x16 tile) = (1-alpha)*I x H(tile) + 0, on the
// matrix pipe. Four chained V_WMMA_F32_16X16X4_F32 with A_j = (1-a)*I[:,4j:4j+4].
// A layout (16x4 f32): lanes0-15 M=0..15 K={0,1}; lanes16-31 K={2,3}.
// B layout (4x16 f32): VGPR0 = rows {0,2}, VGPR1 = rows {1,3} across halves.
// D layout (16x16 f32): VGPR r -> M = r + 8*half, N = lane&15.
// ---------------------------------------------------------------------------
__global__ __launch_bounds__(256) void init_blend_wmma_kernel(
    const float* __restrict__ H, float* __restrict__ Zdst, int n_tiles) {
  int wave = blockIdx.x * (blockDim.x >> 5) + (threadIdx.x >> 5); // wave-uniform
  if (wave >= n_tiles) return;                                   // uniform exit
  int lane = threadIdx.x & 31;
  int half = lane >> 4;        // 0: lanes 0-15, 1: lanes 16-31
  int nn   = lane & 15;        // column / M index within half
  const float s = ONE_MALPHA;

  const float* Hb = H + (size_t)wave * 256;   // 16 nodes x 16 classes
  v8f acc = {};
#pragma unroll
  for (int j = 0; j < 4; ++j) {
    int k0 = half * 2;
    v2f a, b;
    a.x = (nn == j * 4 + k0)     ? s : 0.0f;  // A[m][k0]   = s*delta(m, 4j+k0)
    a.y = (nn == j * 4 + k0 + 1) ? s : 0.0f;  // A[m][k0+1]
    b.x = Hb[(j * 4 + k0)     * 16 + nn];     // B rows 4j+k0, 4j+k0+1
    b.y = Hb[(j * 4 + k0 + 1) * 16 + nn];
    acc = __builtin_amdgcn_wmma_f32_16x16x4_f32(
        /*neg_a=*/false, a, /*neg_b=*/false, b,
        /*c_mod=*/(short)0, acc, /*reuse_a=*/false, /*reuse_b=*/false);
  }

  float* Zb = Zdst + (size_t)wave * 256;
#pragma unroll
  for (int r = 0; r < 8; ++r)
    Zb[(half * 8 + r) * 16 + nn] = acc[r];    // coalesced 64B per half-wave
}

// Tail for N not divisible by 16 (not hit for N=100000, kept for robustness).
__global__ void init_tail_kernel(const float* __restrict__ H,
                                 float* __restrict__ Zdst, int start, int end) {
  int i = start + blockIdx.x * blockDim.x + threadIdx.x;
  if (i < end) Zdst[i] = ONE_MALPHA * H[i];
}

// ---------------------------------------------------------------------------
// Edge scatter: 16 lanes per edge, lane c handles class c.
// Zdst[row,c] += alpha * w_e * Zsrc[col,c].  Gather and atomic bursts are
// 64B-contiguous per edge; wave32 covers 2 edges.
// ---------------------------------------------------------------------------
__global__ __launch_bounds__(256) void edge_scatter_kernel(
    const int* __restrict__ ei, const float* __restrict__ wgt,
    const float* __restrict__ Zsrc, float* __restrict__ Zdst, int n_edges) {
  unsigned gid = blockIdx.x * blockDim.x + threadIdx.x;
  int e = (int)(gid >> 4);
  if (e >= n_edges) return;
  int c   = (int)(gid & 15);
  int row = ei[e];
  int col = ei[n_edges + e];
  float w = wgt[e] * ALPHA_C;
  float v = Zsrc[(size_t)col * 16 + c] * w;
  atomicAdd(Zdst + (size_t)row * 16 + c, v);
}

// ---------------------------------------------------------------------------
extern "C" void kernel_launch(void* const* d_in, const int* in_sizes, int n_in,
                              void* d_out, int out_size, void* d_ws, size_t ws_size,
                              hipStream_t stream) {
  const float* pred    = (const float*)d_in[0];
  const float* margins = (const float*)d_in[1];
  const int*   ei      = (const int*)d_in[2];
  const float* wgt     = (const float*)d_in[3];
  const float* raw     = (const float*)d_in[4];

  const int N_  = in_sizes[1];        // margins: [N]
  const int E_  = in_sizes[3];        // norm_weights: [E]
  const int NL_ = in_sizes[4];        // raw_weights: [NL]

  const size_t NC = (size_t)N_ * 16;
  float* H = (float*)d_ws;            // static injection term
  float* A = H + NC;                  // ping
  float* B = A + NC;                  // pong
  float* out = (float*)d_out;

  const int tB = 256;

  build_h_kernel<<<(N_ + tB - 1) / tB, tB, 0, stream>>>(
      pred, margins, raw, H, A, N_, NL_, NF_CONST);

  const int n_tiles     = N_ / 16;
  const int rem_start   = n_tiles * 16;
  const int init_blocks = (n_tiles + (tB / 32) - 1) / (tB / 32);
  const long long sc_th = (long long)E_ * 16;
  const int sc_blocks   = (int)((sc_th + tB - 1) / tB);

  const float* src = A;
  for (int it = 0; it < NUM_ITER; ++it) {
    float* dst = (it == NUM_ITER - 1) ? out : ((it & 1) ? A : B);
    init_blend_wmma_kernel<<<init_blocks, tB, 0, stream>>>(H, dst, n_tiles);
    if (rem_start < N_) {
      int cnt = N_ * 16 - rem_start * 16;
      init_tail_kernel<<<(cnt + tB - 1) / tB, tB, 0, stream>>>(
          H, dst, rem_start * 16, N_ * 16);
    }
    edge_scatter_kernel<<<sc_blocks, tB, 0, stream>>>(ei, wgt, src, dst, E_);
    src = dst;
  }
}